// GINBlock_39273180954650
// MI455X (gfx1250) — compile-verified
//
#include <hip/hip_runtime.h>
#include <hip/hip_bf16.h>

typedef __attribute__((ext_vector_type(2))) float v2f;
typedef __attribute__((ext_vector_type(8))) float v8f;

#define NNODES   100000
#define NEDGES   1600000
#define DFEAT    64
#define IN_STRIDE 68   // pad so A-frag ds reads (row-striped) hit distinct banks
#define W_STRIDE  72   // pad so B-frag ds reads (lane-halves 2 rows apart) hit distinct banks

// ---------------------------------------------------------------------------
// Zero the aggregation buffer (N*D floats) with float4 stores.
// ---------------------------------------------------------------------------
__global__ __launch_bounds__(256) void zero_kernel(float4* __restrict__ p, int n4) {
    int i = blockIdx.x * 256 + threadIdx.x;
    if (i < n4) p[i] = make_float4(0.f, 0.f, 0.f, 0.f);
}

// ---------------------------------------------------------------------------
// Edge scatter: agg[dst] += h[src].  16 threads per edge, 4 features each.
// Gather of h[src] is a coalesced float4; adds are global_atomic_add_f32
// (RMW lands in L2; the 25.6MB agg array fits in the 192MB L2).
// ---------------------------------------------------------------------------
__global__ __launch_bounds__(256) void scatter_kernel(const float* __restrict__ h,
                                                      const long long* __restrict__ src,
                                                      const long long* __restrict__ dst,
                                                      float* __restrict__ agg) {
    int t = blockIdx.x * 256 + threadIdx.x;           // < E*16 = 25.6M
    if (t >= NEDGES * 16) return;
    int e  = t >> 4;
    int c4 = (t & 15) << 2;
    long long s = src[e];
    long long d = dst[e];
    float4 v = *(const float4*)(h + s * (long long)DFEAT + c4);
    float* ap = agg + d * (long long)DFEAT + c4;
    atomicAdd(ap + 0, v.x);
    atomicAdd(ap + 1, v.y);
    atomicAdd(ap + 2, v.z);
    atomicAdd(ap + 3, v.w);
}

// ---------------------------------------------------------------------------
// Fused GIN MLP layer: out = relu((h+agg) @ W1 + b1) @ W2 + b2  [* mask]
// Block = 128 threads = 4 wave32s; each wave owns a 16-row strip.
// fp32 WMMA 16x16x4: A frag per lane = {Z[row][kb+2*half], Z[row][kb+2*half+1]},
// B frag per lane = {W[kb+2*half][col], W[kb+2*half+1][col]},
// C/D: element i of v8f  ->  row = i + 8*half, col = lane%16.
// ---------------------------------------------------------------------------
__global__ __launch_bounds__(128)
void gin_mlp_kernel(const float* __restrict__ hin, const float* __restrict__ agg,
                    const float* __restrict__ W1, const float* __restrict__ b1,
                    const float* __restrict__ W2, const float* __restrict__ b2,
                    const float* __restrict__ mask, float* __restrict__ out,
                    int applyMask) {
    __shared__ float sIn[64 * IN_STRIDE];
    __shared__ float sW1[64 * W_STRIDE];
    __shared__ float sW2[64 * W_STRIDE];
    __shared__ float sB1[64];
    __shared__ float sB2[64];

    const int tid      = threadIdx.x;
    const int rowbase0 = blockIdx.x * 64;

    // Stage weights (coalesced) and biases into LDS.
    for (int i = tid; i < 64 * 64; i += 128) {
        int r = i >> 6, c = i & 63;
        sW1[r * W_STRIDE + c] = W1[i];
        sW2[r * W_STRIDE + c] = W2[i];
    }
    if (tid < 64) { sB1[tid] = b1[tid]; sB2[tid] = b2[tid]; }

    // Stage Z = h + agg tile (zero-pad rows past N so WMMA runs with EXEC all-1s).
    for (int i = tid; i < 64 * 64; i += 128) {
        int r = i >> 6, c = i & 63;
        int grow = rowbase0 + r;
        float v = 0.f;
        if (grow < NNODES) {
            long long o = (long long)grow * DFEAT + c;
            v = hin[o] + agg[o];
        }
        sIn[r * IN_STRIDE + c] = v;
    }
    __syncthreads();

    const int wave   = tid >> 5;
    const int lane   = tid & 31;
    const int half   = lane >> 4;
    const int lrow   = lane & 15;
    const int rowoff = wave * 16;
    const int arow   = (rowoff + lrow) * IN_STRIDE;

    // ---- GEMM 1: Z @ W1 ----
    v8f acc0 = {}, acc1 = {}, acc2 = {}, acc3 = {};
#pragma unroll
    for (int k = 0; k < 16; ++k) {
        const int kb = 4 * k + 2 * half;
        v2f a;
        a.x = sIn[arow + kb];
        a.y = sIn[arow + kb + 1];
        const float* w0 = &sW1[kb * W_STRIDE + lrow];
        const float* w1 = &sW1[(kb + 1) * W_STRIDE + lrow];
        v2f b;
        b.x = w0[0];  b.y = w1[0];
        acc0 = __builtin_amdgcn_wmma_f32_16x16x4_f32(false, a, false, b, (short)0, acc0, false, false);
        b.x = w0[16]; b.y = w1[16];
        acc1 = __builtin_amdgcn_wmma_f32_16x16x4_f32(false, a, false, b, (short)0, acc1, false, false);
        b.x = w0[32]; b.y = w1[32];
        acc2 = __builtin_amdgcn_wmma_f32_16x16x4_f32(false, a, false, b, (short)0, acc2, false, false);
        b.x = w0[48]; b.y = w1[48];
        acc3 = __builtin_amdgcn_wmma_f32_16x16x4_f32(false, a, false, b, (short)0, acc3, false, false);
    }

    // bias + relu, write back into this wave's own LDS rows (DS in-order per wave,
    // and each wave touches only its private 16-row strip -> no barrier needed).
#pragma unroll
    for (int i = 0; i < 8; ++i) {
        const int r = (rowoff + i + 8 * half) * IN_STRIDE;
        sIn[r + lrow]      = fmaxf(acc0[i] + sB1[lrow],      0.f);
        sIn[r + 16 + lrow] = fmaxf(acc1[i] + sB1[16 + lrow], 0.f);
        sIn[r + 32 + lrow] = fmaxf(acc2[i] + sB1[32 + lrow], 0.f);
        sIn[r + 48 + lrow] = fmaxf(acc3[i] + sB1[48 + lrow], 0.f);
    }

    // ---- GEMM 2: relu(...) @ W2 ----
    acc0 = (v8f){}; acc1 = (v8f){}; acc2 = (v8f){}; acc3 = (v8f){};
#pragma unroll
    for (int k = 0; k < 16; ++k) {
        const int kb = 4 * k + 2 * half;
        v2f a;
        a.x = sIn[arow + kb];
        a.y = sIn[arow + kb + 1];
        const float* w0 = &sW2[kb * W_STRIDE + lrow];
        const float* w1 = &sW2[(kb + 1) * W_STRIDE + lrow];
        v2f b;
        b.x = w0[0];  b.y = w1[0];
        acc0 = __builtin_amdgcn_wmma_f32_16x16x4_f32(false, a, false, b, (short)0, acc0, false, false);
        b.x = w0[16]; b.y = w1[16];
        acc1 = __builtin_amdgcn_wmma_f32_16x16x4_f32(false, a, false, b, (short)0, acc1, false, false);
        b.x = w0[32]; b.y = w1[32];
        acc2 = __builtin_amdgcn_wmma_f32_16x16x4_f32(false, a, false, b, (short)0, acc2, false, false);
        b.x = w0[48]; b.y = w1[48];
        acc3 = __builtin_amdgcn_wmma_f32_16x16x4_f32(false, a, false, b, (short)0, acc3, false, false);
    }

    // bias (+ optional node_mask), store to global.
#pragma unroll
    for (int i = 0; i < 8; ++i) {
        const int grow = rowbase0 + rowoff + i + 8 * half;
        if (grow < NNODES) {
            const float m = applyMask ? mask[grow] : 1.0f;
            float* op = out + (long long)grow * DFEAT;
            op[lrow]      = (acc0[i] + sB2[lrow])      * m;
            op[16 + lrow] = (acc1[i] + sB2[16 + lrow]) * m;
            op[32 + lrow] = (acc2[i] + sB2[32 + lrow]) * m;
            op[48 + lrow] = (acc3[i] + sB2[48 + lrow]) * m;
        }
    }
}

// ---------------------------------------------------------------------------
extern "C" void kernel_launch(void* const* d_in, const int* in_sizes, int n_in,
                              void* d_out, int out_size, void* d_ws, size_t ws_size,
                              hipStream_t stream) {
    const float*     h    = (const float*)d_in[0];
    const float*     x    = (const float*)d_in[1];
    const long long* eidx = (const long long*)d_in[2];     // int64 [2, E]
    const float*     mask = (const float*)d_in[3];
    const float *W1_0 = (const float*)d_in[4],  *b1_0 = (const float*)d_in[5];
    const float *W2_0 = (const float*)d_in[6],  *b2_0 = (const float*)d_in[7];
    const float *W1_1 = (const float*)d_in[8],  *b1_1 = (const float*)d_in[9];
    const float *W2_1 = (const float*)d_in[10], *b2_1 = (const float*)d_in[11];

    const long long* src = eidx;
    const long long* dst = eidx + NEDGES;

    float* agg = (float*)d_ws;                 // N*D floats
    float* h1  = agg + (size_t)NNODES * DFEAT; // N*D floats
    float* out = (float*)d_out;

    const int n4          = NNODES * DFEAT / 4;
    const int zeroBlocks  = (n4 + 255) / 256;
    const int scatBlocks  = (NEDGES * 16 + 255) / 256;
    const int gemmBlocks  = (NNODES + 63) / 64;

    // ---- Layer 0 ----
    zero_kernel<<<zeroBlocks, 256, 0, stream>>>((float4*)agg, n4);
    scatter_kernel<<<scatBlocks, 256, 0, stream>>>(h, src, dst, agg);
    gin_mlp_kernel<<<gemmBlocks, 128, 0, stream>>>(h, agg, W1_0, b1_0, W2_0, b2_0,
                                                   mask, h1, 0);
    // ---- Layer 1 ----
    zero_kernel<<<zeroBlocks, 256, 0, stream>>>((float4*)agg, n4);
    scatter_kernel<<<scatBlocks, 256, 0, stream>>>(h1, src, dst, agg);
    gin_mlp_kernel<<<gemmBlocks, 128, 0, stream>>>(h1, agg, W1_1, b1_1, W2_1, b2_1,
                                                   mask, out, 1);

    // ---- x passthrough: second tuple element, concatenated after h ----
    hipMemcpyAsync(out + (size_t)NNODES * DFEAT, x,
                   (size_t)NNODES * 3 * sizeof(float),
                   hipMemcpyDeviceToDevice, stream);
}